// EDDeform_44195213476583
// MI455X (gfx1250) — compile-verified
//
#include <hip/hip_runtime.h>
#include <hip/hip_bf16.h>

// ---------------------------------------------------------------------------
// EDDeform pipeline for MI455X (gfx1250), fp32 WMMA path.
//   conv0 3x3 (64->64, 130->128)  : implicit GEMM, V_WMMA_F32_16X16X4_F32
//   maxpool 2x2                   : elementwise
//   conv1 1x1 (64->32)            : WMMA GEMM
//   conv2 3x3 pad1 (32->32)       : WMMA implicit GEMM (zero-padded gather)
//   conv3 1x1 (32->1152) offsets  : WMMA GEMM
//   deformable conv (G=64,cpg=1)  : fused bilinear im2col + WMMA GEMM
// All GEMM stages keep full f32 precision (offsets feed floor() in the
// sampler, so no bf16/f16 down-conversion is acceptable).
// Weight staging into LDS uses the CDNA5 one-hop async DMA path
// (GLOBAL_LOAD_ASYNC_TO_LDS_B128, ASYNCcnt) instead of global->VGPR->ds.
// ---------------------------------------------------------------------------

typedef __attribute__((ext_vector_type(2))) float v2f;
typedef __attribute__((ext_vector_type(8))) float v8f;

// Problem constants
#define B_    4
#define CIN   64
#define COUT  64
#define H_    130
#define W_    130
#define HO    128
#define WO    128
#define HP    64          // pooled H/W
#define CMID  32
#define OFFC  1152        // 64 groups * 2 * 9
#define KTOT  576         // 64 * 9
#define KSLAB 144         // 576 / 4 slabs, 64*144*4B = 36 KB LDS per slab

__device__ __forceinline__ v8f wmma4(v2f a, v2f b, v8f c) {
  // D = A(16x4,f32) * B(4x16,f32) + C(16x16,f32)
  return __builtin_amdgcn_wmma_f32_16x16x4_f32(false, a, false, b, (short)0, c,
                                               false, false);
}

// One-hop global->LDS DMA (16B per lane), tracked by ASYNCcnt.
__device__ __forceinline__ void async_copy_b128(unsigned lds_off,
                                                unsigned goff,
                                                const float* base) {
  asm volatile("global_load_async_to_lds_b128 %0, %1, %2"
               :
               : "v"(lds_off), "v"(goff), "s"(base)
               : "memory");
}
__device__ __forceinline__ void wait_async0() {
  asm volatile("s_wait_asynccnt 0x0" ::: "memory");
}

// Stage K-slab s of a (COUT x KTOT) row-major weight matrix into
// wlds (COUT x KSLAB). 2304 float4 chunks, 9 per thread (256 threads).
__device__ __forceinline__ void stage_slab_async(const float* __restrict__ w,
                                                 float* wlds, int s, int tid) {
  unsigned ldsbase = (unsigned)(uintptr_t)wlds;  // low 32b = LDS byte offset
#pragma unroll
  for (int it = 0; it < 9; ++it) {
    int j = tid + 256 * it;        // float4 index within slab
    int e = j * 4;                 // float index within slab
    int m = e / KSLAB;
    int kk = e - m * KSLAB;        // multiple of 4 -> 16B aligned
    unsigned goff = (unsigned)((m * KTOT + s * KSLAB + kk) * 4);
    async_copy_b128(ldsbase + (unsigned)(e * 4), goff, w);
  }
  wait_async0();
}

// ---------------------------------------------------------------------------
// conv0: x(B,64,130,130) * w0(64,64,3,3) -> t0(B,64,128,128)
// Block = 256 thr (8 waves) = one output row (b, ho); wave w owns 16 pixels.
// Each wave accumulates all 64 output channels (4 M-tiles of 16).
// Weights staged in LDS in 4 K-slabs of 144 via async DMA.
// ---------------------------------------------------------------------------
__global__ __launch_bounds__(256) void conv0_wmma(
    const float* __restrict__ x, const float* __restrict__ w,
    const float* __restrict__ bias, float* __restrict__ out) {
  __shared__ float wlds[COUT * KSLAB];  // 36 KB

  const int tid  = threadIdx.x;
  const int ho   = blockIdx.x;
  const int b    = blockIdx.y;
  const int wave = tid >> 5;
  const int lane = tid & 31;
  const int half = lane >> 4;
  const int l    = lane & 15;
  const int wo   = wave * 16 + l;

  const float* xb = x + (size_t)b * CIN * H_ * W_;

  v8f acc[4] = {v8f{}, v8f{}, v8f{}, v8f{}};

  for (int s = 0; s < KTOT / KSLAB; ++s) {
    __syncthreads();                       // protect LDS WAR across slabs
    stage_slab_async(w, wlds, s, tid);     // async DMA + s_wait_asynccnt 0
    __syncthreads();

    for (int k0 = 0; k0 < KSLAB; k0 += 4) {
      // B fragment: im2col rows k, column = pixel (ho, wo)
      v2f bf;
#pragma unroll
      for (int v = 0; v < 2; ++v) {
        int k  = s * KSLAB + k0 + 2 * half + v;
        int c  = k / 9;
        int r  = k % 9;
        int ky = r / 3;
        int kx = r - 3 * ky;
        bf[v] = xb[((size_t)c * H_ + (ho + ky)) * W_ + (wo + kx)];
      }
#pragma unroll
      for (int mt = 0; mt < 4; ++mt) {
        v2f af = *(const v2f*)&wlds[(mt * 16 + l) * KSLAB + k0 + 2 * half];
        acc[mt] = wmma4(af, bf, acc[mt]);
      }
    }
  }

#pragma unroll
  for (int mt = 0; mt < 4; ++mt) {
#pragma unroll
    for (int r = 0; r < 8; ++r) {
      int m = mt * 16 + r + 8 * half;
      out[(((size_t)b * COUT + m) * HO + ho) * WO + wo] = acc[mt][r] + bias[m];
    }
  }
}

// ---------------------------------------------------------------------------
// maxpool 2x2 stride 2: t0(B,64,128,128) -> p(B,64,64,64)
// ---------------------------------------------------------------------------
__global__ __launch_bounds__(256) void maxpool2(const float* __restrict__ in,
                                                float* __restrict__ out) {
  int idx = blockIdx.x * 256 + threadIdx.x;
  const int total = B_ * CIN * HP * HP;
  if (idx >= total) return;
  int j  = idx & 63;
  int i  = (idx >> 6) & 63;
  int bc = idx >> 12;
  const float* p = in + ((size_t)bc * HO + 2 * i) * WO + 2 * j;
  out[idx] = fmaxf(fmaxf(p[0], p[1]), fmaxf(p[WO], p[WO + 1]));
}

// ---------------------------------------------------------------------------
// conv1 1x1: p(B,64,4096) -> t1(B,32,4096).  M=32 (2 tiles), K=64, N=16/wave.
// ---------------------------------------------------------------------------
__global__ __launch_bounds__(256) void conv1_wmma(
    const float* __restrict__ in, const float* __restrict__ w,
    const float* __restrict__ bias, float* __restrict__ out) {
  int gid  = blockIdx.x * 8 + (threadIdx.x >> 5);
  int nt   = gid & 255;
  int b    = gid >> 8;
  int lane = threadIdx.x & 31;
  int half = lane >> 4;
  int l    = lane & 15;
  int n    = nt * 16 + l;

  const float* ib = in + (size_t)b * CIN * (HP * HP);
  v8f acc[2] = {v8f{}, v8f{}};

  for (int k0 = 0; k0 < CIN; k0 += 4) {
    v2f bf;
#pragma unroll
    for (int v = 0; v < 2; ++v)
      bf[v] = ib[(size_t)(k0 + 2 * half + v) * (HP * HP) + n];
#pragma unroll
    for (int mt = 0; mt < 2; ++mt) {
      v2f af = *(const v2f*)&w[(mt * 16 + l) * CIN + k0 + 2 * half];
      acc[mt] = wmma4(af, bf, acc[mt]);
    }
  }
#pragma unroll
  for (int mt = 0; mt < 2; ++mt)
#pragma unroll
    for (int r = 0; r < 8; ++r) {
      int m = mt * 16 + r + 8 * half;
      out[((size_t)b * CMID + m) * (HP * HP) + n] = acc[mt][r] + bias[m];
    }
}

// ---------------------------------------------------------------------------
// conv2 3x3 pad1: t1(B,32,64,64) -> t2(B,32,64,64).  K = 288.
// ---------------------------------------------------------------------------
__global__ __launch_bounds__(256) void conv2_wmma(
    const float* __restrict__ in, const float* __restrict__ w,
    const float* __restrict__ bias, float* __restrict__ out) {
  int gid  = blockIdx.x * 8 + (threadIdx.x >> 5);
  int wt   = gid & 3;
  int h    = (gid >> 2) & 63;
  int b    = gid >> 8;
  int lane = threadIdx.x & 31;
  int half = lane >> 4;
  int l    = lane & 15;
  int wcol = wt * 16 + l;

  const float* ib = in + (size_t)b * CMID * (HP * HP);
  v8f acc[2] = {v8f{}, v8f{}};

  for (int k0 = 0; k0 < CMID * 9; k0 += 4) {
    v2f bf;
#pragma unroll
    for (int v = 0; v < 2; ++v) {
      int k  = k0 + 2 * half + v;
      int c  = k / 9;
      int r  = k % 9;
      int ky = r / 3;
      int kx = r - 3 * ky;
      int iy = h + ky - 1;
      int ix = wcol + kx - 1;
      bf[v] = (iy >= 0 && iy < HP && ix >= 0 && ix < HP)
                  ? ib[((size_t)c * HP + iy) * HP + ix]
                  : 0.0f;
    }
#pragma unroll
    for (int mt = 0; mt < 2; ++mt) {
      v2f af = *(const v2f*)&w[(mt * 16 + l) * (CMID * 9) + k0 + 2 * half];
      acc[mt] = wmma4(af, bf, acc[mt]);
    }
  }
#pragma unroll
  for (int mt = 0; mt < 2; ++mt)
#pragma unroll
    for (int r = 0; r < 8; ++r) {
      int m = mt * 16 + r + 8 * half;
      out[(((size_t)b * CMID + m) * HP + h) * HP + wcol] = acc[mt][r] + bias[m];
    }
}

// ---------------------------------------------------------------------------
// conv3 1x1: t2(B,32,4096) -> offsets t3(B,1152,4096).  M=1152 (72 tiles).
// One wave per 16x16 output tile, K=32.
// ---------------------------------------------------------------------------
__global__ __launch_bounds__(256) void conv3_wmma(
    const float* __restrict__ in, const float* __restrict__ w,
    const float* __restrict__ bias, float* __restrict__ out) {
  int gid  = blockIdx.x * 8 + (threadIdx.x >> 5);
  int nt   = gid & 255;
  int mt   = (gid >> 8) % 72;
  int b    = gid / (256 * 72);
  int lane = threadIdx.x & 31;
  int half = lane >> 4;
  int l    = lane & 15;
  int n    = nt * 16 + l;

  const float* ib = in + (size_t)b * CMID * (HP * HP);
  v8f acc = v8f{};

  for (int k0 = 0; k0 < CMID; k0 += 4) {
    v2f bf;
#pragma unroll
    for (int v = 0; v < 2; ++v)
      bf[v] = ib[(size_t)(k0 + 2 * half + v) * (HP * HP) + n];
    v2f af = *(const v2f*)&w[(mt * 16 + l) * CMID + k0 + 2 * half];
    acc = wmma4(af, bf, acc);
  }
#pragma unroll
  for (int r = 0; r < 8; ++r) {
    int m = mt * 16 + r + 8 * half;
    out[((size_t)b * OFFC + m) * (HP * HP) + n] = acc[r] + bias[m];
  }
}

// ---------------------------------------------------------------------------
// Deformable conv, fused bilinear im2col + WMMA GEMM.
// offsets t3(B,1152,64,64) used at (ho>>1, wo>>1)  (== repeat(2) upsample).
// Group g == channel c (cpg = 1): off_y ch = c*18 + kk*2, off_x = +1.
// out(B,64,128,128) = wd(64,576) @ cols(576, pixel).
// ---------------------------------------------------------------------------
__device__ __forceinline__ float tap(const float* __restrict__ xc, int iy,
                                     int ix) {
  return (iy >= 0 && iy < H_ && ix >= 0 && ix < W_) ? xc[iy * W_ + ix] : 0.0f;
}

__global__ __launch_bounds__(256) void deform_wmma(
    const float* __restrict__ x, const float* __restrict__ offs,
    const float* __restrict__ wd, float* __restrict__ out) {
  __shared__ float wlds[COUT * KSLAB];  // 36 KB

  const int tid  = threadIdx.x;
  const int ho   = blockIdx.x;
  const int b    = blockIdx.y;
  const int wave = tid >> 5;
  const int lane = tid & 31;
  const int half = lane >> 4;
  const int l    = lane & 15;
  const int wo   = wave * 16 + l;
  const int oh   = ho >> 1;
  const int ow   = wo >> 1;

  const float* xb = x + (size_t)b * CIN * H_ * W_;
  const float* ob = offs + (size_t)b * OFFC * (HP * HP);

  v8f acc[4] = {v8f{}, v8f{}, v8f{}, v8f{}};

  for (int s = 0; s < KTOT / KSLAB; ++s) {
    __syncthreads();
    stage_slab_async(wd, wlds, s, tid);    // async DMA + s_wait_asynccnt 0
    __syncthreads();

    for (int k0 = 0; k0 < KSLAB; k0 += 4) {
      v2f bf;
#pragma unroll
      for (int v = 0; v < 2; ++v) {
        int k  = s * KSLAB + k0 + 2 * half + v;
        int c  = k / 9;
        int kk = k % 9;
        int ky = kk / 3;
        int kx = kk - 3 * ky;
        float oy = ob[(size_t)(c * 18 + kk * 2) * (HP * HP) + oh * HP + ow];
        float ox = ob[(size_t)(c * 18 + kk * 2 + 1) * (HP * HP) + oh * HP + ow];
        float py = oy + (float)(ho + ky);
        float px = ox + (float)(wo + kx);
        float y0 = floorf(py);
        float x0 = floorf(px);
        float fy = py - y0;
        float fx = px - x0;
        int iy = (int)y0;
        int ix = (int)x0;
        const float* xc = xb + (size_t)c * H_ * W_;
        float v00 = tap(xc, iy, ix);
        float v01 = tap(xc, iy, ix + 1);
        float v10 = tap(xc, iy + 1, ix);
        float v11 = tap(xc, iy + 1, ix + 1);
        bf[v] = (1.0f - fy) * ((1.0f - fx) * v00 + fx * v01) +
                fy * ((1.0f - fx) * v10 + fx * v11);
      }
#pragma unroll
      for (int mt = 0; mt < 4; ++mt) {
        v2f af = *(const v2f*)&wlds[(mt * 16 + l) * KSLAB + k0 + 2 * half];
        acc[mt] = wmma4(af, bf, acc[mt]);
      }
    }
  }

#pragma unroll
  for (int mt = 0; mt < 4; ++mt) {
#pragma unroll
    for (int r = 0; r < 8; ++r) {
      int m = mt * 16 + r + 8 * half;
      out[(((size_t)b * COUT + m) * HO + ho) * WO + wo] = acc[mt][r];
    }
  }
}

// ---------------------------------------------------------------------------
extern "C" void kernel_launch(void* const* d_in, const int* in_sizes, int n_in,
                              void* d_out, int out_size, void* d_ws,
                              size_t ws_size, hipStream_t stream) {
  (void)in_sizes; (void)n_in; (void)out_size; (void)ws_size;

  const float* x  = (const float*)d_in[0];
  const float* w0 = (const float*)d_in[1];
  const float* b0 = (const float*)d_in[2];
  const float* w1 = (const float*)d_in[3];
  const float* b1 = (const float*)d_in[4];
  const float* w2 = (const float*)d_in[5];
  const float* b2 = (const float*)d_in[6];
  const float* w3 = (const float*)d_in[7];
  const float* b3 = (const float*)d_in[8];
  const float* wd = (const float*)d_in[9];
  float* out = (float*)d_out;

  // Workspace layout (floats): 25,165,824 total = 96 MB
  float* t0 = (float*)d_ws;          // B*64*128*128 = 4,194,304
  float* pp = t0 + 4194304;          // B*64*64*64   = 1,048,576
  float* t1 = pp + 1048576;          // B*32*64*64   =   524,288
  float* t2 = t1 + 524288;           // B*32*64*64   =   524,288
  float* t3 = t2 + 524288;           // B*1152*64*64 = 18,874,368

  conv0_wmma<<<dim3(HO, B_), 256, 0, stream>>>(x, w0, b0, t0);
  maxpool2<<<(B_ * CIN * HP * HP + 255) / 256, 256, 0, stream>>>(t0, pp);
  conv1_wmma<<<(B_ * 256) / 8, 256, 0, stream>>>(pp, w1, b1, t1);
  conv2_wmma<<<(B_ * 64 * 4) / 8, 256, 0, stream>>>(t1, w2, b2, t2);
  conv3_wmma<<<(B_ * 72 * 256) / 8, 256, 0, stream>>>(t2, w3, b3, t3);
  deform_wmma<<<dim3(HO, B_), 256, 0, stream>>>(x, t3, wd, out);
}